// DistortionLoss_82471962018601
// MI455X (gfx1250) — compile-verified
//
#include <hip/hip_runtime.h>

typedef float v2f __attribute__((ext_vector_type(2)));
typedef float v8f __attribute__((ext_vector_type(8)));

#define WAVES_PER_BLOCK 2
#define RAYS_PER_WAVE   16
#define NSAMP           128

// Distortion loss via sorted-midpoint prefix trick, prefix computed as a
// triangular matmul-scan on V_WMMA_F32_16X16X4_F32 (wave32, 16 rays/wave).
__global__ __launch_bounds__(WAVES_PER_BLOCK * 32)
void DistortionLoss_82471962018601_kernel(const float* __restrict__ wp,
                                          const float* __restrict__ zp,
                                          const float* __restrict__ nearp,
                                          const float* __restrict__ farp,
                                          float* __restrict__ out,
                                          int nrays) {
    // LDS: per wave, a = w*m and b = w stored [sample][ray] (sample-major),
    // plus the per-ray third-term scalars.
    __shared__ float sA[WAVES_PER_BLOCK][NSAMP * RAYS_PER_WAVE];
    __shared__ float sB[WAVES_PER_BLOCK][NSAMP * RAYS_PER_WAVE];
    __shared__ float sT3[WAVES_PER_BLOCK][RAYS_PER_WAVE];

    const int wave  = threadIdx.x >> 5;
    const int lane  = threadIdx.x & 31;
    const int rbase = blockIdx.x * (WAVES_PER_BLOCK * RAYS_PER_WAVE) + wave * RAYS_PER_WAVE;

    // ---------------- Phase 0: load + preprocess into LDS ----------------
    for (int r = 0; r < RAYS_PER_WAVE; ++r) {
        int ray = rbase + r;
        if (ray >= nrays) ray = nrays - 1;          // clamp (R divisible anyway)
        const float nv  = nearp[ray];
        const float fv  = farp[ray];
        const float inv = 1.0f / (fv - nv);
        float t3 = 0.0f;
#pragma unroll
        for (int q = 0; q < 4; ++q) {
            const int i   = q * 32 + lane;          // sample index 0..127
            const float w = wp[(long)ray * NSAMP + i];
            const float zlo = zp[(long)ray * (NSAMP + 1) + i];
            const float zhi = zp[(long)ray * (NSAMP + 1) + i + 1];
            const float blo = (zlo - nv) * inv;
            const float bhi = (zhi - nv) * inv;
            const float m   = 0.5f * (blo + bhi);
            sB[wave][i * RAYS_PER_WAVE + r] = w;
            sA[wave][i * RAYS_PER_WAVE + r] = w * m;
            t3 += w * w * (bhi - blo);
        }
        // full-wave reduction of third term for this ray
        for (int off = 16; off > 0; off >>= 1) t3 += __shfl_xor(t3, off, 32);
        if (lane == 0) sT3[wave][r] = t3;
    }
    __syncthreads();

    // ---------------- Phase 1: matmul-scan + dot, per 16-sample chunk ----
    const int ray16  = lane & 15;    // ray column owned by this lane (B/C/D layout)
    const int hi     = lane >> 4;    // 0: K/M rows {0,1}/{g}, 1: rows {2,3}/{g+8}
    const int rowOff = hi * 8;
    const int srow   = lane & 15;    // A-matrix row M for the S fragment

    // Strict-lower-triangular S fragments for the 4 K-steps (16x4 A layout:
    // vgpr0 holds K = 4*kk + 2*hi, vgpr1 holds K = 4*kk + 2*hi + 1).
    v2f sfrag[4];
#pragma unroll
    for (int kk = 0; kk < 4; ++kk) {
        const int col0 = kk * 4 + hi * 2;
        sfrag[kk].x = (col0     < srow) ? 1.0f : 0.0f;
        sfrag[kk].y = (col0 + 1 < srow) ? 1.0f : 0.0f;
    }

    float carryA = 0.0f, carryB = 0.0f;  // exclusive cross-chunk prefix (per ray)
    float dot = 0.0f;

    for (int c = 0; c < 8; ++c) {
        const float* baseB = &sB[wave][(c * 16) * RAYS_PER_WAVE];
        const float* baseA = &sA[wave][(c * 16) * RAYS_PER_WAVE];

        v8f Db = {}; // exclusive in-chunk prefix of b, rows x rays
        v8f Da = {}; // exclusive in-chunk prefix of a
#pragma unroll
        for (int kk = 0; kk < 4; ++kk) {
            // B fragment: K-rows (4kk + 2*hi, +1), column = ray16
            const int kr = kk * 4 + hi * 2;
            v2f bf, af;
            bf.x = baseB[(kr    ) * RAYS_PER_WAVE + ray16];
            bf.y = baseB[(kr + 1) * RAYS_PER_WAVE + ray16];
            af.x = baseA[(kr    ) * RAYS_PER_WAVE + ray16];
            af.y = baseA[(kr + 1) * RAYS_PER_WAVE + ray16];
            Db = __builtin_amdgcn_wmma_f32_16x16x4_f32(
                     false, sfrag[kk], false, bf, (short)0, Db, false, false);
            Da = __builtin_amdgcn_wmma_f32_16x16x4_f32(
                     false, sfrag[kk], false, af, (short)0, Da, false, false);
        }

        // Dot against a/b for the 8 rows this lane owns, and chunk sums.
        float sb = 0.0f, sa = 0.0f;
#pragma unroll
        for (int g = 0; g < 8; ++g) {
            const float bv = baseB[(g + rowOff) * RAYS_PER_WAVE + ray16];
            const float av = baseA[(g + rowOff) * RAYS_PER_WAVE + ray16];
            dot += av * (Db[g] + carryB) - bv * (Da[g] + carryA);
            sb += bv;
            sa += av;
        }
        sb += __shfl_xor(sb, 16, 32);   // combine both row-halves -> full chunk sum
        sa += __shfl_xor(sa, 16, 32);
        carryB += sb;
        carryA += sa;
    }

    dot += __shfl_xor(dot, 16, 32);     // combine row halves of the dot product

    if (lane < 16 && (rbase + lane) < nrays) {
        out[rbase + lane] = 2.0f * dot + (1.0f / 3.0f) * sT3[wave][lane];
    }
}

extern "C" void kernel_launch(void* const* d_in, const int* in_sizes, int n_in,
                              void* d_out, int out_size, void* d_ws, size_t ws_size,
                              hipStream_t stream) {
    const float* wp    = (const float*)d_in[0]; // weights (R,128,1)
    const float* zp    = (const float*)d_in[1]; // z_vals  (R,129)
    const float* nearp = (const float*)d_in[2]; // near    (R,1)
    const float* farp  = (const float*)d_in[3]; // far     (R,1)
    float* out = (float*)d_out;                 // loss    (R,1)

    const int nrays = in_sizes[2];              // R = 8192
    const int raysPerBlock = WAVES_PER_BLOCK * RAYS_PER_WAVE; // 32
    const int blocks = (nrays + raysPerBlock - 1) / raysPerBlock;

    DistortionLoss_82471962018601_kernel<<<blocks, WAVES_PER_BLOCK * 32, 0, stream>>>(
        wp, zp, nearp, farp, out, nrays);
}